// SemiSupervisedGraphSAGE_43499428774650
// MI455X (gfx1250) — compile-verified
//
#include <hip/hip_runtime.h>

// ---------------------------------------------------------------------------
// GraphSAGE forward for MI455X (gfx1250, wave32).
//   layer: neigh = scatter_mean(x[src] -> dst); y = ReLU([x|neigh] @ W.T + b)
//   head : logits = Wc2 @ ReLU(Wc1 @ x2 + bc1) + bc2
// Scatter is L2-atomic-bound (features fit in 192MB L2); the GEMMs are tiny,
// so we run them on the fp32 WMMA path (16x16x4) to keep full fp32 precision.
// ---------------------------------------------------------------------------

#define GN_NODES 100000
#define GN_EDGES 1600000
#define GD       64

typedef float v2f __attribute__((ext_vector_type(2)));
typedef float v8f __attribute__((ext_vector_type(8)));

// ---------------------------------------------------------------- degree ----
__global__ __launch_bounds__(256) void degree_kernel(const int* __restrict__ dst,
                                                     float* __restrict__ deg, int E) {
  int e = blockIdx.x * 256 + threadIdx.x;
  if (e < E) atomicAdd(&deg[dst[e]], 1.0f);
}

__global__ __launch_bounds__(256) void invdeg_kernel(float* __restrict__ deg, int n) {
  int i = blockIdx.x * 256 + threadIdx.x;
  if (i < n) deg[i] = 1.0f / fmaxf(deg[i], 1.0f);
}

// -------------------------------------------------------------- scatter -----
// 64 consecutive threads handle the 64 dims of one edge: gather of x[src] and
// atomic adds into acc[dst] are whole-cacheline coalesced; edge id is
// wave-uniform so the index loads scalarize.
__global__ __launch_bounds__(256) void scatter_kernel(const float* __restrict__ x,
                                                      const int* __restrict__ src,
                                                      const int* __restrict__ dst,
                                                      float* __restrict__ acc, int E) {
  unsigned tid = blockIdx.x * 256u + threadIdx.x;
  int e = (int)(tid >> 6);
  int j = (int)(tid & 63u);
  if (e < E) {
    int s = src[e];
    int d = dst[e];
    atomicAdd(&acc[d * GD + j], x[s * GD + j]);
  }
}

// ---------------------------------------------------- SAGE layer via WMMA ---
// One wave owns a 16-node x 64-col output tile:
//   out[16,64] = ReLU([x | neigh*invdeg](16x128) @ W.T(128x64) + b)
// A tile staged in LDS (coalesced, deg-normalized at stage time); K marched in
// chunks of 4 with V_WMMA_F32_16X16X4_F32 (fp32 in, fp32 acc = exact path).
// Block = 128 threads = 4 waves = 64 nodes; 32 KB LDS.
__global__ __launch_bounds__(128) void sage_gemm_wmma(const float* __restrict__ x,
                                                      const float* __restrict__ neigh,
                                                      const float* __restrict__ invdeg,
                                                      const float* __restrict__ W,
                                                      const float* __restrict__ bias,
                                                      float* __restrict__ out, int n) {
  __shared__ float smem[4 * 16 * 128];  // 4 waves x (16 rows x 128 cols)
  const int lane = threadIdx.x & 31;
  const int wave = threadIdx.x >> 5;
  const int node0 = (blockIdx.x * 4 + wave) * 16;
  float* my = &smem[wave * 16 * 128];

  // Stage self features into cols [0,64)
#pragma unroll 8
  for (int i = 0; i < 32; ++i) {
    int idx = i * 32 + lane;
    int row = idx >> 6, col = idx & 63;
    int node = node0 + row;
    if (node >= n) node = n - 1;  // tail clamp keeps EXEC uniform for WMMA
    my[row * 128 + col] = x[node * GD + col];
  }
  // Stage mean-aggregated neighbor features into cols [64,128)
#pragma unroll 8
  for (int i = 0; i < 32; ++i) {
    int idx = i * 32 + lane;
    int row = idx >> 6, col = idx & 63;
    int node = node0 + row;
    if (node >= n) node = n - 1;
    my[row * 128 + 64 + col] = neigh[node * GD + col] * invdeg[node];
  }
  __syncthreads();

  const v8f vzero = {0.f, 0.f, 0.f, 0.f, 0.f, 0.f, 0.f, 0.f};
  v8f acc[4];
#pragma unroll
  for (int t = 0; t < 4; ++t) acc[t] = vzero;

  // A frag (16x4 f32): lanes 0-15 carry K={kb,kb+1}, lanes 16-31 K={kb+2,kb+3}
  const int arow  = lane & 15;
  const int khalf = (lane >> 4) << 1;
  const int ncol  = lane & 15;

#pragma unroll
  for (int kc = 0; kc < 32; ++kc) {
    const int kb = (kc << 2) + khalf;
    v2f a;
    a.x = my[arow * 128 + kb];
    a.y = my[arow * 128 + kb + 1];
#pragma unroll
    for (int nt = 0; nt < 4; ++nt) {
      // B = W.T chunk: B[k][ncol] = W[(nt*16+ncol)*128 + k]
      const float* wp = &W[(nt * 16 + ncol) * 128 + kb];
      v2f bfrag;
      bfrag.x = wp[0];
      bfrag.y = wp[1];
      acc[nt] = __builtin_amdgcn_wmma_f32_16x16x4_f32(
          /*neg_a=*/false, a, /*neg_b=*/false, bfrag,
          /*c_mod=*/(short)0, acc[nt], /*reuse_a=*/false, /*reuse_b=*/false);
    }
  }

  // C/D layout: VGPR v -> M = v + 8*(lane>>4); N = 16*nt + (lane&15)
  const int mbase = (lane >> 4) << 3;
#pragma unroll
  for (int nt = 0; nt < 4; ++nt) {
    const int col = nt * 16 + ncol;
    const float bv = bias[col];
#pragma unroll
    for (int v = 0; v < 8; ++v) {
      int node = node0 + mbase + v;
      float val = fmaxf(acc[nt][v] + bv, 0.0f);
      if (node < n) out[node * GD + col] = val;
    }
  }
}

// ------------------------------------------------------------- classifier ---
// h = ReLU(Wc1(32x64) @ x + bc1); logits = Wc2(2x32) @ h + bc2.
// Tiny (0.4 GFLOP total) -> VALU FMAs with weights cached in LDS.
__global__ __launch_bounds__(256) void classifier_kernel(const float* __restrict__ x,
                                                         const float* __restrict__ Wc1,
                                                         const float* __restrict__ bc1,
                                                         const float* __restrict__ Wc2,
                                                         const float* __restrict__ bc2,
                                                         float* __restrict__ out, int n) {
  __shared__ float sW1[32 * 64];
  __shared__ float sb1[32];
  __shared__ float sW2[2 * 32];
  __shared__ float sb2[2];
  const int t = threadIdx.x;
  for (int i = t; i < 32 * 64; i += 256) sW1[i] = Wc1[i];
  if (t < 32) sb1[t] = bc1[t];
  if (t < 64) sW2[t] = Wc2[t];
  if (t < 2) sb2[t] = bc2[t];
  __syncthreads();

  const int node = blockIdx.x * 256 + t;
  if (node >= n) return;

  float xr[64];
  const float4* xp = (const float4*)(x + (size_t)node * GD);
#pragma unroll
  for (int i = 0; i < 16; ++i) {
    float4 v = xp[i];
    xr[4 * i + 0] = v.x; xr[4 * i + 1] = v.y;
    xr[4 * i + 2] = v.z; xr[4 * i + 3] = v.w;
  }

  float h[32];
#pragma unroll 4
  for (int k = 0; k < 32; ++k) {
    float s = sb1[k];
#pragma unroll
    for (int j = 0; j < 64; ++j) s = fmaf(sW1[k * 64 + j], xr[j], s);
    h[k] = fmaxf(s, 0.0f);
  }
#pragma unroll
  for (int c = 0; c < 2; ++c) {
    float s = sb2[c];
#pragma unroll
    for (int k = 0; k < 32; ++k) s = fmaf(sW2[c * 32 + k], h[k], s);
    out[node * 2 + c] = s;
  }
}

// ------------------------------------------------------------------ driver --
extern "C" void kernel_launch(void* const* d_in, const int* in_sizes, int n_in,
                              void* d_out, int out_size, void* d_ws, size_t ws_size,
                              hipStream_t stream) {
  const float* features = (const float*)d_in[0];
  const int*   ei       = (const int*)d_in[1];  // int64 in source degrades to i32 (JAX x64 off)
  const float* W1  = (const float*)d_in[2];
  const float* b1  = (const float*)d_in[3];
  const float* W2  = (const float*)d_in[4];
  const float* b2  = (const float*)d_in[5];
  const float* Wc1 = (const float*)d_in[6];
  const float* bc1 = (const float*)d_in[7];
  const float* Wc2 = (const float*)d_in[8];
  const float* bc2 = (const float*)d_in[9];
  float* out = (float*)d_out;

  const int N = GN_NODES;
  const int E = GN_EDGES;
  const int* src = ei;
  const int* dst = ei + E;

  // Workspace layout (~77 MB): invdeg | neigh accumulator | x1 | x2
  float* deg   = (float*)d_ws;
  float* neigh = deg + N;
  float* x1    = neigh + (size_t)N * GD;
  float* x2    = x1 + (size_t)N * GD;

  const int edgeBlocks    = (E + 255) / 256;
  const int nodeBlocks    = (N + 255) / 256;
  const int scatterBlocks = (int)(((size_t)E * GD + 255) / 256);  // 400000
  const int gemmBlocks    = (N + 63) / 64;                        // 4 tiles/block

  // Degrees -> 1/max(deg,1), shared by both layers.
  hipMemsetAsync(deg, 0, (size_t)N * sizeof(float), stream);
  degree_kernel<<<edgeBlocks, 256, 0, stream>>>(dst, deg, E);
  invdeg_kernel<<<nodeBlocks, 256, 0, stream>>>(deg, N);

  // ---- Layer 1 ----
  hipMemsetAsync(neigh, 0, (size_t)N * GD * sizeof(float), stream);
  scatter_kernel<<<scatterBlocks, 256, 0, stream>>>(features, src, dst, neigh, E);
  sage_gemm_wmma<<<gemmBlocks, 128, 0, stream>>>(features, neigh, deg, W1, b1, x1, N);

  // ---- Layer 2 ----
  hipMemsetAsync(neigh, 0, (size_t)N * GD * sizeof(float), stream);
  scatter_kernel<<<scatterBlocks, 256, 0, stream>>>(x1, src, dst, neigh, E);
  sage_gemm_wmma<<<gemmBlocks, 128, 0, stream>>>(x1, neigh, deg, W2, b2, x2, N);

  // ---- Head ----
  classifier_kernel<<<nodeBlocks, 256, 0, stream>>>(x2, Wc1, bc1, Wc2, bc2, out, N);
}